// MatchDistance_20177756356663
// MI455X (gfx1250) — compile-verified
//
#include <hip/hip_runtime.h>
#include <stdint.h>

// -------- CDNA5 (gfx1250) fused match-distance pipeline --------
// B=32, NZ=NM=FIN=128, FQK=64.  Output d: (524288, 128) f32 = 256MB.
// Bandwidth-bound by the output stream -> f32 WMMA for all GEMMs (exact),
// async global->LDS staging (ASYNCcnt), NT b128 stores for d,
// ds_add_f32 for the softmax denominator.

typedef __attribute__((ext_vector_type(2))) float v2f;
typedef __attribute__((ext_vector_type(4))) float v4f;
typedef __attribute__((ext_vector_type(8))) float v8f;
typedef __attribute__((ext_vector_type(4))) int   v4i;

#define B_    32
#define NZ_   128
#define NM_   128
#define FIN_  128
#define FQK_  64
#define NEDGE (B_ * NZ_ * NM_)       /* 524288 */
#define GHALF ((uint32_t)NEDGE)      /* n/2 with n = 2*NEDGE Gumbel draws */

#if defined(__HIP_DEVICE_COMPILE__)
# if __has_builtin(__builtin_amdgcn_wmma_f32_16x16x4_f32)
#  define HAVE_WMMA_F32X4 1
# else
#  define HAVE_WMMA_F32X4 0
#  warning "gfx1250: f32 WMMA builtin unavailable on device pass - scalar FMA fallback"
# endif
# if __has_builtin(__builtin_amdgcn_global_load_async_to_lds_b128)
#  define HAVE_ASYNC_LDS 1
# else
#  define HAVE_ASYNC_LDS 0
#  warning "gfx1250: global_load_async_to_lds_b128 builtin unavailable - VGPR staging fallback"
# endif
#else
# define HAVE_WMMA_F32X4 0
# define HAVE_ASYNC_LDS 0
#endif

#define AS1 __attribute__((address_space(1)))
#define AS3 __attribute__((address_space(3)))

// ---- async global->LDS b128 copy (GLOBAL_LOAD_ASYNC_TO_LDS_B128, ASYNCcnt) ----
// builtin signature (from hipcc diagnostic): param0 = AS1 v4i* (global src),
// param1 = LDS dest, imm offset, imm cpol.
__device__ __forceinline__ void async_copy_b128(float* lds_dst, const float* gsrc) {
#if HAVE_ASYNC_LDS
  __builtin_amdgcn_global_load_async_to_lds_b128(
      (AS1 v4i*)gsrc, (AS3 v4i*)lds_dst, /*imm offset*/0, /*cpol*/0);
#else
  *(v4f*)lds_dst = *(const v4f*)gsrc;
#endif
}

__device__ __forceinline__ void async_wait_all() {
#if HAVE_ASYNC_LDS
# if __has_builtin(__builtin_amdgcn_s_wait_asynccnt)
  __builtin_amdgcn_s_wait_asynccnt(0);
# else
  asm volatile("s_wait_asynccnt 0x0" ::: "memory");
# endif
#endif
}

// ---------------- Threefry2x32 (JAX PRNG, key = jax.random.key(42) = {0,42}) ----------------
__device__ __forceinline__ uint32_t rotl32(uint32_t x, uint32_t n) {
  return (x << n) | (x >> (32u - n));
}

__device__ __forceinline__ void threefry2x32(uint32_t x0, uint32_t x1,
                                             uint32_t& o0, uint32_t& o1) {
  const uint32_t k0 = 0u, k1 = 42u;
  const uint32_t k2 = 0x1BD11BDAu ^ k0 ^ k1;
  const uint32_t ks[3] = {k0, k1, k2};
  const uint32_t rotA[4] = {13u, 15u, 26u, 6u};
  const uint32_t rotB[4] = {17u, 29u, 16u, 24u};
  x0 += k0; x1 += k1;
#pragma unroll
  for (int i = 0; i < 5; ++i) {
    const uint32_t* rot = (i & 1) ? rotB : rotA;
#pragma unroll
    for (int r = 0; r < 4; ++r) { x0 += x1; x1 = rotl32(x1, rot[r]); x1 ^= x0; }
    x0 += ks[(i + 1) % 3];
    x1 += ks[(i + 2) % 3] + (uint32_t)(i + 1);
  }
  o0 = x0; o1 = x1;
}

// JAX random_bits for even count n: halves of iota are the two threefry inputs.
__device__ __forceinline__ uint32_t random_bits_at(uint32_t p) {
  uint32_t o0, o1;
  if (p < GHALF) { threefry2x32(p, p + GHALF, o0, o1); return o0; }
  threefry2x32(p - GHALF, p, o0, o1); return o1;
}

// jax.random.uniform(minval=1e-10, maxval=1.0) -> gumbel -log(-log(u))
__device__ __forceinline__ float gumbel_from_bits(uint32_t bits) {
  float f = __uint_as_float((bits >> 9) | 0x3f800000u) - 1.0f;   // [0,1)
  float u = fmaxf(1e-10f, fmaf(f, 1.0f - 1e-10f, 1e-10f));
  return -__logf(-__logf(u));
}

// ---------------- Kernel 1: q/k projection via f32 WMMA ----------------
// grid 512: blocks [0,256) -> q tiles from z, [256,512) -> k tiles from m.
// Each block: 16 output rows x 64 cols; 4 waves, one 16x16 col-tile each.
__global__ __launch_bounds__(128, 1)
void md_proj_wmma(const float* __restrict__ z, const float* __restrict__ m,
                  const float* __restrict__ W,
                  float* __restrict__ q_ws, float* __restrict__ k_ws) {
  __shared__ float A_lds[16][132];   // 16 rows x FIN, pad 132 (bank stride 4)
  __shared__ float W_lds[128][64];   // K x N chunk of W

  const int tid  = threadIdx.x;
  const int lane = tid & 31;         // wave32
  const int wave = tid >> 5;
  const bool isK = blockIdx.x >= 256;
  const int tile = blockIdx.x & 255;

  const float* src  = isK ? m : z;
  float*       dst  = isK ? k_ws : q_ws;
  const int    c0   = isK ? FQK_ : 0;
  const float  scl  = isK ? 1.0f : 0.125f;   // FQK^-0.5 folded into q

  for (int i = tid; i < 16 * 32; i += 128) {          // 16x128 A tile (b128)
    int r = i >> 5, c4 = i & 31;
    async_copy_b128(&A_lds[r][c4 * 4],
                    src + ((size_t)(tile * 16 + r)) * FIN_ + c4 * 4);
  }
  for (int i = tid; i < 128 * 16; i += 128) {         // 128x64 W chunk (b128)
    int r = i >> 4, c4 = i & 15;
    async_copy_b128(&W_lds[r][c4 * 4],
                    W + (size_t)r * (2 * FQK_) + c0 + c4 * 4);
  }
  async_wait_all();
  __syncthreads();

  const int n0  = wave * 16;
  const int l15 = lane & 15;
  const int hi  = lane >> 4;
  const int kh  = hi << 1;
  v8f acc = {};

#if HAVE_WMMA_F32X4
#pragma unroll
  for (int k0 = 0; k0 < FIN_; k0 += 4) {
    v2f a, b;
    a.x = A_lds[l15][k0 + kh];       a.y = A_lds[l15][k0 + kh + 1];
    b.x = W_lds[k0 + kh][n0 + l15];  b.y = W_lds[k0 + kh + 1][n0 + l15];
    acc = __builtin_amdgcn_wmma_f32_16x16x4_f32(false, a, false, b,
                                                (short)0, acc, false, false);
  }
#else
#pragma unroll
  for (int r = 0; r < 8; ++r) {
    int row = r + (hi << 3);
    float s = 0.f;
    for (int k = 0; k < FIN_; ++k) s = fmaf(A_lds[row][k], W_lds[k][n0 + l15], s);
    acc[r] = s;
  }
#endif

#pragma unroll
  for (int r = 0; r < 8; ++r) {                       // C layout: M=r (+8 hi-half)
    int row = tile * 16 + r + (hi << 3);
    dst[(size_t)row * FQK_ + n0 + l15] = acc[r] * scl;
  }
}

// ---------------- Kernel 2: scores + softmax + gumbel + streamed d ----------------
// grid 256 = (batch, 16-z-row tile); 256 threads = 8 waves; wave w owns
// the 16 m-columns [16w, 16w+16) of the 16x128 score tile.
__global__ __launch_bounds__(256, 1)
void md_fused(const float* __restrict__ q_ws, const float* __restrict__ k_ws,
              const float* __restrict__ z, const float* __restrict__ m,
              float* __restrict__ d) {
  __shared__ float k_lds[NM_][68];    // pad 68: stride-68 col gathers conflict-free
  __shared__ float q_lds[16][68];
  __shared__ float z_lds[16][FIN_];
  __shared__ float s_lds[16][NM_];    // exp(scores)
  __shared__ float w_lds[16][NM_];    // hard gumbel weights
  __shared__ float denom[16];

  const int tid  = threadIdx.x;
  const int lane = tid & 31;
  const int wave = tid >> 5;
  const int bb   = blockIdx.x >> 3;
  const int zi0  = (blockIdx.x & 7) * 16;

  for (int i = tid; i < 16 * 16; i += 256) {          // q tile 16x64 (b128)
    int r = i >> 4, c4 = i & 15;
    async_copy_b128(&q_lds[r][c4 * 4],
                    q_ws + ((size_t)(bb * NZ_ + zi0 + r)) * FQK_ + c4 * 4);
  }
  for (int i = tid; i < 128 * 16; i += 256) {         // k batch 128x64 (b128)
    int r = i >> 4, c4 = i & 15;
    async_copy_b128(&k_lds[r][c4 * 4],
                    k_ws + ((size_t)(bb * NM_ + r)) * FQK_ + c4 * 4);
  }
  for (int i = tid; i < 16 * 32; i += 256) {          // z tile 16x128 (b128)
    int r = i >> 5, c4 = i & 31;
    async_copy_b128(&z_lds[r][c4 * 4],
                    z + ((size_t)(bb * NZ_ + zi0 + r)) * FIN_ + c4 * 4);
  }
  async_wait_all();
  __syncthreads();

  // ---- S = q @ k^T (16x16 tile per wave), f32 WMMA K-loop over 64 ----
  const int n0  = wave * 16;
  const int l15 = lane & 15;
  const int hi  = lane >> 4;
  const int kh  = hi << 1;
  v8f acc = {};

#if HAVE_WMMA_F32X4
#pragma unroll
  for (int k0 = 0; k0 < FQK_; k0 += 4) {
    v2f a, b;
    a.x = q_lds[l15][k0 + kh];       a.y = q_lds[l15][k0 + kh + 1];
    b.x = k_lds[n0 + l15][k0 + kh];  b.y = k_lds[n0 + l15][k0 + kh + 1];  // k^T
    acc = __builtin_amdgcn_wmma_f32_16x16x4_f32(false, a, false, b,
                                                (short)0, acc, false, false);
  }
#else
#pragma unroll
  for (int r = 0; r < 8; ++r) {
    int row = r + (hi << 3);
    float s = 0.f;
    for (int k = 0; k < FQK_; ++k) s = fmaf(q_lds[row][k], k_lds[n0 + l15][k], s);
    acc[r] = s;
  }
#endif

#pragma unroll
  for (int r = 0; r < 8; ++r)
    s_lds[r + (hi << 3)][n0 + l15] = __expf(acc[r]);
  __syncthreads();

  if (tid < 16) denom[tid] = 0.f;
  __syncthreads();
  {
    int row = tid >> 4, cc = (tid & 15) * 8;
    float s = 0.f;
#pragma unroll
    for (int c = 0; c < 8; ++c) s += s_lds[row][cc + c];
    atomicAdd(&denom[row], s);                        // ds_add_f32
  }
  __syncthreads();

  // ---- Gumbel straight-through hard weight (exact JAX Threefry) ----
#pragma unroll
  for (int i = 0; i < 8; ++i) {
    int el = tid * 8 + i;                             // 0..2047
    int r = el >> 7, j = el & 127;
    float aw = s_lds[r][j] / denom[r];
    uint32_t e = (uint32_t)((bb * NZ_ + zi0 + r) * NM_ + j);
    float g0 = gumbel_from_bits(random_bits_at(2u * e));
    float g1 = gumbel_from_bits(random_bits_at(2u * e + 1u));
    // argmax over [(aw+g0)/TAU, (1-aw+g1)/TAU]; first index wins ties
    w_lds[r][j] = ((aw + g0) >= (1.0f - aw + g1)) ? 1.0f : 0.0f;
  }
  __syncthreads();

  // ---- stream d = (z - m)^2 * w : 2048 edges x 128 feats, NT b128 stores ----
  const size_t base = ((size_t)(bb * NZ_ + zi0)) << 14;     // * NM_ * FIN_
  for (int idx = tid; idx < 16 * 128 * 32; idx += 256) {
    int el = idx >> 5;                 // one edge per wave-iteration
    int f4 = idx & 31;                 // lane -> contiguous float4 in feature dim
    int r = el >> 7, j = el & 127;
    float w = w_lds[r][j];
    v4f zv = *(const v4f*)&z_lds[r][f4 * 4];
    v4f mv = *(const v4f*)(m + ((size_t)(bb * NM_ + j)) * FIN_ + f4 * 4);  // L2-hot
    v4f df = zv - mv;
    v4f out = df * df * w;
    __builtin_nontemporal_store(out, (v4f*)(d + base + ((size_t)el << 7) + f4 * 4));
  }
}

// ---------------- launch ----------------
extern "C" void kernel_launch(void* const* d_in, const int* in_sizes, int n_in,
                              void* d_out, int out_size, void* d_ws, size_t ws_size,
                              hipStream_t stream) {
  (void)in_sizes; (void)n_in; (void)out_size; (void)ws_size;
  const float* z = (const float*)d_in[0];   // (32,128,128)
  const float* m = (const float*)d_in[1];   // (32,128,128)
  const float* W = (const float*)d_in[2];   // (128,128)
  float* d = (float*)d_out;                 // (524288,128)
  float* q_ws = (float*)d_ws;               // 4096x64 f32 = 1MB
  float* k_ws = q_ws + (size_t)(B_ * NZ_) * FQK_;  // +1MB

  md_proj_wmma<<<dim3(512), dim3(128), 0, stream>>>(z, m, W, q_ws, k_ws);
  md_fused<<<dim3(256), dim3(256), 0, stream>>>(q_ws, k_ws, z, m, d);
}